// NeuralODE_8117488190111
// MI455X (gfx1250) — compile-verified
//
#include <hip/hip_runtime.h>
#include <hip/hip_bf16.h>
#include <cstddef>

// ---------------------------------------------------------------------------
// Persistent Tsit5 Neural-ODE integrator for MI455X (gfx1250, wave32).
// One workgroup owns 16 batch rows and integrates the full trajectory.
// All three MLP GEMMs run on V_WMMA_F32_16X16X4_F32 (native fp32 matrix core)
// to match the fp32 reference precision across 1080 chained GEMMs.
// Weights stay resident in the 192MB L2; activations/k-stages live in LDS.
// ---------------------------------------------------------------------------

typedef __attribute__((ext_vector_type(2))) float v2f;
typedef __attribute__((ext_vector_type(8))) float v8f;

#define LDY 68    // padded row stride for [16 x 64] buffers (bank-conflict free)
#define LDH 516   // padded row stride for [16 x 512] buffers
#define NROW 16

// Tsit5 tableau (5th-order weights, b7 = 0)
#define A21f 0.161f
#define A31f (-0.008480655492356989f)
#define A32f 0.335480655492357f
#define A41f 2.8971530571054935f
#define A42f (-6.359448489975075f)
#define A43f 4.3622954328695815f
#define A51f 5.325864828439257f
#define A52f (-11.748883564062828f)
#define A53f 7.4955393428898365f
#define A54f (-0.09249506636175525f)
#define A61f 5.86145544294642f
#define A62f (-12.92096931784711f)
#define A63f 8.159367898576159f
#define A64f (-0.071584973281401f)
#define A65f (-0.028269050394068383f)
#define B1f 0.09646076681806523f
#define B2f 0.01f
#define B3f 0.4798896504144996f
#define B4f 1.379008574103742f
#define B5f (-3.290069515436081f)
#define B6f 2.324710524099774f

__device__ __forceinline__ v8f wmma4(v2f a, v2f b, v8f c) {
  // D = A(16x4,f32) * B(4x16,f32) + C(16x16,f32)
  return __builtin_amdgcn_wmma_f32_16x16x4_f32(
      /*neg_a=*/false, a, /*neg_b=*/false, b,
      /*c_mod=*/(short)0, c, /*reuse_a=*/false, /*reuse_b=*/false);
}

__device__ __forceinline__ float softplus_f(float x) {
  // numerically stable log1p(exp(x))
  return fmaxf(x, 0.0f) + log1pf(__expf(-fabsf(x)));
}

// Hdst[16][512] = softplus( Asrc[16][K] @ Wg[512][K]^T + bias[512] )
// 8 waves x 4 N-tiles each; A fragment (LDS) reused across the 4 tiles.
__device__ __forceinline__ void gemm_act(
    const float* __restrict__ Asrc, int lda, int K,
    const float* __restrict__ Wg, int ldw,
    const float* __restrict__ bias,
    float* __restrict__ Hdst,
    int lane, int wave)
{
  const int half = lane >> 4;        // K sub-block select (lanes 16-31 -> K+2)
  const int l15  = lane & 15;        // M for A, N for B
  const int nb   = wave * 64 + l15;  // W row of this wave's first N-tile

  v8f acc0 = {}, acc1 = {}, acc2 = {}, acc3 = {};
  const float* ap = Asrc + l15 * lda + half * 2;
  const float* wp = Wg + (size_t)nb * ldw + half * 2;

#pragma unroll 4
  for (int k = 0; k < K; k += 4) {
    v2f av;  av.x  = ap[k];              av.y  = ap[k + 1];
    v2f bv0; bv0.x = wp[k];              bv0.y = wp[k + 1];
    v2f bv1; bv1.x = wp[16 * ldw + k];   bv1.y = wp[16 * ldw + k + 1];
    v2f bv2; bv2.x = wp[32 * ldw + k];   bv2.y = wp[32 * ldw + k + 1];
    v2f bv3; bv3.x = wp[48 * ldw + k];   bv3.y = wp[48 * ldw + k + 1];
    acc0 = wmma4(av, bv0, acc0);
    acc1 = wmma4(av, bv1, acc1);
    acc2 = wmma4(av, bv2, acc2);
    acc3 = wmma4(av, bv3, acc3);
  }

#pragma unroll
  for (int r = 0; r < 8; ++r) {
    const int m = r + half * 8;
    float* hp = Hdst + m * LDH + nb;
    hp[0]  = softplus_f(acc0[r] + bias[nb]);
    hp[16] = softplus_f(acc1[r] + bias[nb + 16]);
    hp[32] = softplus_f(acc2[r] + bias[nb + 32]);
    hp[48] = softplus_f(acc3[r] + bias[nb + 48]);
  }
}

// k-partials[16][64] = Asrc[16][512] @ W2[64][512]^T (+ b2 on half 0)
// 8 waves = 4 N-tiles x 2 K-halves; partials reduced in the combine pass.
__device__ __forceinline__ void gemm_out(
    const float* __restrict__ Asrc,
    const float* __restrict__ W2, const float* __restrict__ b2,
    float* __restrict__ kp0, float* __restrict__ kp1,
    int lane, int wave)
{
  const int half  = lane >> 4;
  const int l15   = lane & 15;
  const int ntile = wave & 3;
  const int khalf = wave >> 2;
  const int n     = ntile * 16 + l15;

  v8f acc = {};
  const float* ap = Asrc + l15 * LDH + khalf * 256 + half * 2;
  const float* wp = W2 + (size_t)n * 512 + khalf * 256 + half * 2;

#pragma unroll 4
  for (int k = 0; k < 256; k += 4) {
    v2f av; av.x = ap[k]; av.y = ap[k + 1];
    v2f bv; bv.x = wp[k]; bv.y = wp[k + 1];
    acc = wmma4(av, bv, acc);
  }

  float* kp = khalf ? kp1 : kp0;
  const float badd = khalf ? 0.0f : b2[n];
#pragma unroll
  for (int r = 0; r < 8; ++r) {
    const int m = r + half * 8;
    kp[m * LDY + n] = acc[r] + badd;
  }
}

// One vector-field eval: ysrc[16][64] (LDS) -> kp0+kp1 partials (LDS)
__device__ __forceinline__ void feval(
    const float* __restrict__ ysrc,
    float* __restrict__ H1, float* __restrict__ H2,
    float* __restrict__ kp0, float* __restrict__ kp1,
    const float* __restrict__ W0, const float* __restrict__ b0,
    const float* __restrict__ W1, const float* __restrict__ b1,
    const float* __restrict__ W2, const float* __restrict__ b2,
    int lane, int wave)
{
  gemm_act(ysrc, LDY, 64,  W0, 64,  b0, H1, lane, wave);
  __syncthreads();
  gemm_act(H1,   LDH, 512, W1, 512, b1, H2, lane, wave);
  __syncthreads();
  gemm_out(H2, W2, b2, kp0, kp1, lane, wave);
  __syncthreads();
}

// LDS layout (floats): 9 * 16*68 + 2 * 16*516 = 26304 floats = 105216 bytes
#define SMEM_FLOATS (9 * NROW * LDY + 2 * NROW * LDH)

__global__ __launch_bounds__(256, 1) void node_tsit5_kernel(
    const float* __restrict__ ts, const float* __restrict__ y0,
    const float* __restrict__ W0, const float* __restrict__ b0,
    const float* __restrict__ W1, const float* __restrict__ b1,
    const float* __restrict__ W2, const float* __restrict__ b2,
    float* __restrict__ out)
{
  extern __shared__ float smem[];
  float* yin  = smem;                  // current state   [16][LDY]
  float* ytmp = yin  + NROW * LDY;     // stage input     [16][LDY]
  float* kb0  = ytmp + NROW * LDY;     // k1..k5 storage
  float* kb1  = kb0  + NROW * LDY;
  float* kb2  = kb1  + NROW * LDY;
  float* kb3  = kb2  + NROW * LDY;
  float* kb4  = kb3  + NROW * LDY;
  float* kp0  = kb4  + NROW * LDY;     // GEMM3 K-half partials
  float* kp1  = kp0  + NROW * LDY;
  float* H1   = kp1  + NROW * LDY;     // hidden 1 [16][LDH]
  float* H2   = H1   + NROW * LDH;     // hidden 2 [16][LDH]

  const int tid  = threadIdx.x;
  const int lane = tid & 31;
  const int wave = tid >> 5;
  const int row0 = blockIdx.x * NROW;  // 32 blocks x 16 rows = B=512

  // Load y0 slice, emit t=0 output. out is [B][T=16][D=64].
  for (int e = tid; e < NROW * 64; e += 256) {
    const int r = e >> 6, d = e & 63;
    const float v = y0[(size_t)(row0 + r) * 64 + d];
    yin[r * LDY + d] = v;
    out[(size_t)(row0 + r) * (16 * 64) + d] = v;
  }
  __syncthreads();

  for (int t = 0; t < 15; ++t) {
    const float h = (ts[t + 1] - ts[t]) * 0.25f;  // SUBSTEPS = 4

    for (int ss = 0; ss < 4; ++ss) {
      // ---- stage 1: k1 = f(y) ----
      feval(yin, H1, H2, kp0, kp1, W0, b0, W1, b1, W2, b2, lane, wave);
      for (int e = tid; e < NROW * 64; e += 256) {
        const int o = ((e >> 6) * LDY) + (e & 63);
        const float k1 = kp0[o] + kp1[o];
        kb0[o] = k1;
        ytmp[o] = yin[o] + h * (A21f * k1);
      }
      __syncthreads();

      // ---- stage 2 ----
      feval(ytmp, H1, H2, kp0, kp1, W0, b0, W1, b1, W2, b2, lane, wave);
      for (int e = tid; e < NROW * 64; e += 256) {
        const int o = ((e >> 6) * LDY) + (e & 63);
        const float k2 = kp0[o] + kp1[o];
        kb1[o] = k2;
        ytmp[o] = yin[o] + h * (A31f * kb0[o] + A32f * k2);
      }
      __syncthreads();

      // ---- stage 3 ----
      feval(ytmp, H1, H2, kp0, kp1, W0, b0, W1, b1, W2, b2, lane, wave);
      for (int e = tid; e < NROW * 64; e += 256) {
        const int o = ((e >> 6) * LDY) + (e & 63);
        const float k3 = kp0[o] + kp1[o];
        kb2[o] = k3;
        ytmp[o] = yin[o] + h * (A41f * kb0[o] + A42f * kb1[o] + A43f * k3);
      }
      __syncthreads();

      // ---- stage 4 ----
      feval(ytmp, H1, H2, kp0, kp1, W0, b0, W1, b1, W2, b2, lane, wave);
      for (int e = tid; e < NROW * 64; e += 256) {
        const int o = ((e >> 6) * LDY) + (e & 63);
        const float k4 = kp0[o] + kp1[o];
        kb3[o] = k4;
        ytmp[o] = yin[o] + h * (A51f * kb0[o] + A52f * kb1[o] +
                                A53f * kb2[o] + A54f * k4);
      }
      __syncthreads();

      // ---- stage 5 ----
      feval(ytmp, H1, H2, kp0, kp1, W0, b0, W1, b1, W2, b2, lane, wave);
      for (int e = tid; e < NROW * 64; e += 256) {
        const int o = ((e >> 6) * LDY) + (e & 63);
        const float k5 = kp0[o] + kp1[o];
        kb4[o] = k5;
        ytmp[o] = yin[o] + h * (A61f * kb0[o] + A62f * kb1[o] +
                                A63f * kb2[o] + A64f * kb3[o] + A65f * k5);
      }
      __syncthreads();

      // ---- stage 6 + 5th-order update ----
      feval(ytmp, H1, H2, kp0, kp1, W0, b0, W1, b1, W2, b2, lane, wave);
      for (int e = tid; e < NROW * 64; e += 256) {
        const int o = ((e >> 6) * LDY) + (e & 63);
        const float k6 = kp0[o] + kp1[o];
        yin[o] = yin[o] + h * (B1f * kb0[o] + B2f * kb1[o] + B3f * kb2[o] +
                               B4f * kb3[o] + B5f * kb4[o] + B6f * k6);
      }
      __syncthreads();
    }

    // save y at t+1
    for (int e = tid; e < NROW * 64; e += 256) {
      const int r = e >> 6, d = e & 63;
      out[(size_t)(row0 + r) * (16 * 64) + (size_t)(t + 1) * 64 + d] =
          yin[r * LDY + d];
    }
    __syncthreads();
  }
}

extern "C" void kernel_launch(void* const* d_in, const int* in_sizes, int n_in,
                              void* d_out, int out_size, void* d_ws, size_t ws_size,
                              hipStream_t stream) {
  const float* ts = (const float*)d_in[0];
  const float* y0 = (const float*)d_in[1];
  const float* W0 = (const float*)d_in[2];
  const float* b0 = (const float*)d_in[3];
  const float* W1 = (const float*)d_in[4];
  const float* b1 = (const float*)d_in[5];
  const float* W2 = (const float*)d_in[6];
  const float* b2 = (const float*)d_in[7];
  float* out = (float*)d_out;

  const size_t shmem = (size_t)SMEM_FLOATS * sizeof(float);
  node_tsit5_kernel<<<32, 256, shmem, stream>>>(ts, y0, W0, b0, W1, b1, W2, b2, out);
}